// NeuralMemory_24068996726879
// MI455X (gfx1250) — compile-verified
//
#include <hip/hip_runtime.h>

#define DEVI __device__ __forceinline__

typedef __attribute__((ext_vector_type(2))) float v2f;
typedef __attribute__((ext_vector_type(8))) float v8f;

constexpr int Bn = 2, Ln = 256, Dn = 128, Hn = 256;
constexpr int LD = Ln * Dn;   // 32768
constexpr int LH = Ln * Hn;   // 65536
constexpr int LL = Ln * Ln;   // 65536
// per-batch workspace layout (float offsets)
constexpr int OQ = 0, OKk = LD, OV = 2 * LD;
constexpr int OX2 = 3 * LD, OSB1 = OX2 + LH, OGZ2 = OSB1 + LH;
constexpr int OGZ1 = OGZ2 + LD, OU1 = OGZ1 + LH, OXQ2 = OU1 + LH;
constexpr int OU2 = OXQ2 + LH, OS1 = OU2 + LD, OS2 = OS1 + LL;
constexpr int OSC = OS2 + LL;                 // lr | cw | cm | Te | ee  (5*L)
constexpr int PB  = OSC + 5 * Ln;             // per-batch stride in floats

DEVI float sigmoidf_(float z) { return 1.0f / (1.0f + __expf(-z)); }
DEVI float siluf_(float z)    { return z * sigmoidf_(z); }
DEVI float silubwdf_(float z) { float sg = sigmoidf_(z); float s = z * sg; return s + sg * (1.0f - s); }
DEVI float softplusf_(float z){ return fmaxf(z, 0.0f) + log1pf(__expf(-fabsf(z))); }

DEVI v8f wmma4(v2f a, v2f b, v8f c) {
  // V_WMMA_F32_16X16X4_F32 : D = A(16x4,f32) * B(4x16,f32) + C(16x16,f32)
  return __builtin_amdgcn_wmma_f32_16x16x4_f32(
      /*neg_a=*/false, a, /*neg_b=*/false, b,
      /*c_mod=*/(short)0, c, /*reuse_a=*/false, /*reuse_b=*/false);
}

// C(16x16) = A(16xK, row-major, lda) * B(16xK, row-major, ldb)^T  -- "TN"
// A is pre-offset to its row0, Bt pre-offset to its col0 (as rows of Bt).
DEVI v8f tile_tn(const float* __restrict__ A, int lda,
                 const float* __restrict__ Bt, int ldb, int K) {
  const int lane = threadIdx.x & 31;
  const int m  = lane & 15;
  const int kk = (lane >> 4) << 1;            // K sub-offset per ISA A/B layout
  const float* ap = A  + m * lda + kk;
  const float* bp = Bt + m * ldb + kk;
  v8f acc = {};
#pragma unroll 8
  for (int k = 0; k < K; k += 4) {
    v2f a = *(const v2f*)(ap + k);
    v2f b = *(const v2f*)(bp + k);
    acc = wmma4(a, b, acc);
  }
  return acc;
}

// C(16x16) = A(16xK, row-major, lda) * B(Kx16, row-major, ldb)  -- "NN"
// A pre-offset to row0, B pre-offset to col0.
DEVI v8f tile_nn(const float* __restrict__ A, int lda,
                 const float* __restrict__ B, int ldb, int K) {
  const int lane = threadIdx.x & 31;
  const int m  = lane & 15;
  const int kk = (lane >> 4) << 1;
  const float* ap = A + m * lda + kk;
  const float* bp = B + kk * ldb + m;
  v8f acc = {};
#pragma unroll 8
  for (int k = 0; k < K; k += 4) {
    v2f a = *(const v2f*)(ap + k);
    v2f b;
    b.x = bp[k * ldb];
    b.y = bp[k * ldb + ldb];
    acc = wmma4(a, b, acc);
  }
  return acc;
}

// ---------- 1) q,k,v projections:  [L x 3D] = x(LxD) @ [Wq;Wk;Wv]^T + bias ----------
__global__ void k_qkv(const float* __restrict__ x,
                      const float* __restrict__ Wq, const float* __restrict__ bq,
                      const float* __restrict__ Wk, const float* __restrict__ bk,
                      const float* __restrict__ Wv, const float* __restrict__ bv,
                      float* __restrict__ ws) {
  const int b = blockIdx.z;
  const int row0 = blockIdx.x * 16;
  const int col  = blockIdx.y * 16;            // 0..383
  const int which = col / Dn;
  const int coff  = col - which * Dn;
  const float* W    = (which == 0) ? Wq : (which == 1) ? Wk : Wv;
  const float* bias = (which == 0) ? bq : (which == 1) ? bk : bv;
  float* dst = ws + (size_t)b * PB + (size_t)which * LD;
  const float* A = x + (size_t)b * LD + row0 * Dn;

  v8f acc = tile_tn(A, Dn, W + coff * Dn, Dn, Dn);

  const int lane = threadIdx.x & 31;
  const int rb = row0 + ((lane >> 4) << 3);
  const int cl = coff + (lane & 15);
  const float bs = bias[cl];
#pragma unroll
  for (int j = 0; j < 8; ++j)
    dst[(rb + j) * Dn + cl] = acc[j] + bs;
}

// ---------- 2) per-token scalars + scans: lr, cw, cm, Te, ee ----------
__global__ void k_scan(const float* __restrict__ x,
                       const float* __restrict__ Wlr, const float* __restrict__ blr,
                       const float* __restrict__ Wm,  const float* __restrict__ bm,
                       const float* __restrict__ Wwd, const float* __restrict__ bwd,
                       float* __restrict__ ws) {
  const int b = blockIdx.x;
  const int l = threadIdx.x;                   // 0..255
  __shared__ float sm[Ln], sw[Ln], se[Ln];

  const float* xv = x + ((size_t)b * Ln + l) * Dn;
  float zl = blr[0], zm = bm[0], zw = bwd[0];
#pragma unroll 4
  for (int d = 0; d < Dn; ++d) {
    float xd = xv[d];
    zl = fmaf(xd, Wlr[d], zl);
    zm = fmaf(xd, Wm[d],  zm);
    zw = fmaf(xd, Wwd[d], zw);
  }
  const float lrv = softplusf_(zl);
  sm[l] = -softplusf_(-zm);                    // log_mom
  sw[l] = -softplusf_(-zw);                    // log_wd
  __syncthreads();
  for (int off = 1; off < Ln; off <<= 1) {     // inclusive Hillis-Steele scans
    float tm = (l >= off) ? sm[l - off] : 0.0f;
    float tw = (l >= off) ? sw[l - off] : 0.0f;
    __syncthreads();
    sm[l] += tm; sw[l] += tw;
    __syncthreads();
  }
  const float cmv = sm[l], cwv = sw[l];
  const float ev = __expf(cmv - cwv);
  se[l] = ev;
  __syncthreads();
  for (int off = 1; off < Ln; off <<= 1) {
    float t = (l >= off) ? se[l - off] : 0.0f;
    __syncthreads();
    se[l] += t;
    __syncthreads();
  }
  float* sc = ws + (size_t)b * PB + OSC;
  sc[l]          = lrv;
  sc[Ln + l]     = cwv;
  sc[2 * Ln + l] = cmv;
  sc[3 * Ln + l] = se[l];                      // Te (inclusive prefix of ee)
  sc[4 * Ln + l] = ev;                         // ee
}

// ---------- 3/6) [L x H] = src(LxD) @ W1^T + b1; mode0: k -> X2,SB1 ; mode1: q -> U1 ----------
__global__ void k_h16(float* __restrict__ ws, const float* __restrict__ W1,
                      const float* __restrict__ b1, int mode) {
  const int b = blockIdx.z;
  float* wsb = ws + (size_t)b * PB;
  const int row0 = blockIdx.x * 16;
  const int col0 = blockIdx.y * 16;
  const float* A = wsb + (mode ? OQ : OKk) + row0 * Dn;

  v8f acc = tile_tn(A, Dn, W1 + col0 * Dn, Dn, Dn);

  const int lane = threadIdx.x & 31;
  const int rb = row0 + ((lane >> 4) << 3);
  const int hc = col0 + (lane & 15);
  const float bs = b1[hc];
#pragma unroll
  for (int j = 0; j < 8; ++j) {
    float z = acc[j] + bs;
    if (mode == 0) {
      wsb[OX2  + (rb + j) * Hn + hc] = siluf_(z);
      wsb[OSB1 + (rb + j) * Hn + hc] = silubwdf_(z);
    } else {
      wsb[OU1 + (rb + j) * Hn + hc] = z;
    }
  }
}

// ---------- 4/9) [L x D] = src(LxH) @ W2^T + b2; mode0: X2 -> gZ2=Z2-v ; mode1: Xq2 -> U2 ----------
__global__ void k_d16(float* __restrict__ ws, const float* __restrict__ W2,
                      const float* __restrict__ b2, int mode) {
  const int b = blockIdx.z;
  float* wsb = ws + (size_t)b * PB;
  const int row0 = blockIdx.x * 16;
  const int col0 = blockIdx.y * 16;
  const float* A = wsb + (mode ? OXQ2 : OX2) + row0 * Hn;

  v8f acc = tile_tn(A, Hn, W2 + col0 * Hn, Hn, Hn);

  const int lane = threadIdx.x & 31;
  const int rb = row0 + ((lane >> 4) << 3);
  const int dc = col0 + (lane & 15);
  const float bs = b2[dc];
#pragma unroll
  for (int j = 0; j < 8; ++j) {
    float z = acc[j] + bs;
    if (mode == 0)
      wsb[OGZ2 + (rb + j) * Dn + dc] = z - wsb[OV + (rb + j) * Dn + dc];
    else
      wsb[OU2 + (rb + j) * Dn + dc] = z;
  }
}

// ---------- 5) gZ1 = (gZ2(LxD) @ W2(DxH)) * silu_bwd(Z1) ----------
__global__ void k_gz1(float* __restrict__ ws, const float* __restrict__ W2) {
  const int b = blockIdx.z;
  float* wsb = ws + (size_t)b * PB;
  const int row0 = blockIdx.x * 16;
  const int col0 = blockIdx.y * 16;

  v8f acc = tile_nn(wsb + OGZ2 + row0 * Dn, Dn, W2 + col0, Hn, Dn);

  const int lane = threadIdx.x & 31;
  const int rb = row0 + ((lane >> 4) << 3);
  const int hc = col0 + (lane & 15);
#pragma unroll
  for (int j = 0; j < 8; ++j)
    wsb[OGZ1 + (rb + j) * Hn + hc] = acc[j] * wsb[OSB1 + (rb + j) * Hn + hc];
}

// ---------- 7/10) masked score matrices: S[l,m] = A[l,m]*lr[m]*(P[l,m]+1), m<=l ----------
// phase0: P = q @ k^T (K=D) -> S1 ; phase1: P = Xq2 @ X2^T (K=H) -> S2
__global__ void k_scores(float* __restrict__ ws, int phase) {
  const int b = blockIdx.z;
  float* wsb = ws + (size_t)b * PB;
  const int row0 = blockIdx.x * 16;
  const int col0 = blockIdx.y * 16;
  const int K  = phase ? Hn : Dn;              // also the leading dim of both operands
  const float* Abuf = wsb + (phase ? OXQ2 : OQ);
  const float* Bbuf = wsb + (phase ? OX2  : OKk);
  float* Sdst = wsb + (phase ? OS2 : OS1);
  const float* sc = wsb + OSC;

  v8f acc = tile_tn(Abuf + row0 * K, K, Bbuf + col0 * K, K, K);

  const int lane = threadIdx.x & 31;
  const int rb = row0 + ((lane >> 4) << 3);
  const int m  = col0 + (lane & 15);
  const float lrm = sc[m];
  const float cmm = sc[2 * Ln + m];
  const float Tem = sc[3 * Ln + m];
  const float eem = sc[4 * Ln + m];
#pragma unroll
  for (int j = 0; j < 8; ++j) {
    const int l = rb + j;
    float s = 0.0f;
    if (m <= l) {
      // A[l,m] = exp(cw[l]-cm[m]) * (Te[l] - Te[m-1]),  Te[m-1] = Te[m]-ee[m]
      float Ae = __expf(sc[Ln + l] - cmm) * (sc[3 * Ln + l] - Tem + eem);
      s = Ae * lrm * (acc[j] + 1.0f);
    }
    Sdst[l * Ln + m] = s;
  }
}

// ---------- 8) Xq2 = silu( S1(LxL) @ gZ1(LxH) + exp(cw[l]) * U1 ) ----------
__global__ void k_zq1(float* __restrict__ ws) {
  const int b = blockIdx.z;
  float* wsb = ws + (size_t)b * PB;
  const int row0 = blockIdx.x * 16;
  const int col0 = blockIdx.y * 16;
  const float* sc = wsb + OSC;

  v8f acc = tile_nn(wsb + OS1 + row0 * Ln, Ln, wsb + OGZ1 + col0, Hn, Ln);

  const int lane = threadIdx.x & 31;
  const int rb = row0 + ((lane >> 4) << 3);
  const int hc = col0 + (lane & 15);
#pragma unroll
  for (int j = 0; j < 8; ++j) {
    const int l = rb + j;
    float z = acc[j] + __expf(sc[Ln + l]) * wsb[OU1 + l * Hn + hc];
    wsb[OXQ2 + l * Hn + hc] = siluf_(z);
  }
}

// ---------- 11) out = S2(LxL) @ gZ2(LxD) + exp(cw[l]) * U2 ----------
__global__ void k_out(const float* __restrict__ ws, float* __restrict__ out) {
  const int b = blockIdx.z;
  const float* wsb = ws + (size_t)b * PB;
  const int row0 = blockIdx.x * 16;
  const int col0 = blockIdx.y * 16;
  const float* sc = wsb + OSC;

  v8f acc = tile_nn(wsb + OS2 + row0 * Ln, Ln, wsb + OGZ2 + col0, Dn, Ln);

  const int lane = threadIdx.x & 31;
  const int rb = row0 + ((lane >> 4) << 3);
  const int dc = col0 + (lane & 15);
#pragma unroll
  for (int j = 0; j < 8; ++j) {
    const int l = rb + j;
    out[((size_t)b * Ln + l) * Dn + dc] =
        acc[j] + __expf(sc[Ln + l]) * wsb[OU2 + l * Dn + dc];
  }
}

extern "C" void kernel_launch(void* const* d_in, const int* in_sizes, int n_in,
                              void* d_out, int out_size, void* d_ws, size_t ws_size,
                              hipStream_t stream) {
  const float* x   = (const float*)d_in[0];
  const float* W1  = (const float*)d_in[1];
  const float* b1  = (const float*)d_in[2];
  const float* W2  = (const float*)d_in[3];
  const float* b2  = (const float*)d_in[4];
  const float* Wq  = (const float*)d_in[5];
  const float* bq  = (const float*)d_in[6];
  const float* Wk  = (const float*)d_in[7];
  const float* bk  = (const float*)d_in[8];
  const float* Wv  = (const float*)d_in[9];
  const float* bv  = (const float*)d_in[10];
  const float* Wlr = (const float*)d_in[11];
  const float* blr = (const float*)d_in[12];
  const float* Wm  = (const float*)d_in[13];
  const float* bm  = (const float*)d_in[14];
  const float* Wwd = (const float*)d_in[15];
  const float* bwd = (const float*)d_in[16];
  float* out = (float*)d_out;
  float* ws  = (float*)d_ws;

  k_qkv   <<<dim3(Ln/16, 3*Dn/16, Bn), 32, 0, stream>>>(x, Wq, bq, Wk, bk, Wv, bv, ws);
  k_scan  <<<dim3(Bn), Ln, 0, stream>>>(x, Wlr, blr, Wm, bm, Wwd, bwd, ws);
  k_h16   <<<dim3(Ln/16, Hn/16, Bn), 32, 0, stream>>>(ws, W1, b1, 0);   // Z1 -> X2, SB1
  k_d16   <<<dim3(Ln/16, Dn/16, Bn), 32, 0, stream>>>(ws, W2, b2, 0);   // gZ2 = Z2 - v
  k_gz1   <<<dim3(Ln/16, Hn/16, Bn), 32, 0, stream>>>(ws, W2);          // gZ1
  k_h16   <<<dim3(Ln/16, Hn/16, Bn), 32, 0, stream>>>(ws, W1, b1, 1);   // U1
  k_scores<<<dim3(Ln/16, Ln/16, Bn), 32, 0, stream>>>(ws, 0);           // S1
  k_zq1   <<<dim3(Ln/16, Hn/16, Bn), 32, 0, stream>>>(ws);              // Xq2
  k_d16   <<<dim3(Ln/16, Dn/16, Bn), 32, 0, stream>>>(ws, W2, b2, 1);   // U2
  k_scores<<<dim3(Ln/16, Ln/16, Bn), 32, 0, stream>>>(ws, 1);           // S2
  k_out   <<<dim3(Ln/16, Dn/16, Bn), 32, 0, stream>>>(ws, out);         // Zq2
}